// HGTModelOptimized_37855841747405
// MI455X (gfx1250) — compile-verified
//
#include <hip/hip_runtime.h>
#include <hip/hip_bf16.h>
#include <math.h>

// ---------------------------------------------------------------------------
// HGT forward for MI455X (gfx1250, wave32).
//  * All [N,128]x[128,128] GEMMs on the matrix pipe via V_WMMA_F32_16X16X4_F32
//    (f32 in / f32 acc -> matches reference numerics).
//  * Weights consumed TRANSPOSED (Wt[col][k]) so each B fragment is one
//    global_load_b64; 32 rows/block so each B fragment feeds two WMMAs.
//  * Relation transforms a_rel/m_rel folded into kW/vW on device
//    (W' = W * blockdiag(R)), written transposed directly.
//  * Edge softmax: encoded-uint atomicMax for segment max, f32 HW atomics
//    (unsafeAtomicAdd -> global_atomic_add_f32) for denom + message scatter.
// Scratch requirement: ~191 MB.
// ---------------------------------------------------------------------------

#define ND 20000
#define NG 40000
#define HID 128
#define LAYERS 2
#define RPB 32          // GEMM rows per block; 20000%32==0 && 40000%32==0

typedef float v2f __attribute__((ext_vector_type(2)));
typedef float v8f __attribute__((ext_vector_type(8)));

// ---- monotone encoding of f32 into u32 for atomicMax-based segment max ----
__device__ __forceinline__ unsigned encf(float f) {
  unsigned u = __float_as_uint(f);
  return (u & 0x80000000u) ? ~u : (u | 0x80000000u);
}
__device__ __forceinline__ float decf(unsigned u) {
  u = (u & 0x80000000u) ? (u & 0x7FFFFFFFu) : ~u;
  return __uint_as_float(u);
}
#define ENC_NEGINF 0x007FFFFFu  // encf(-inf)

__device__ __forceinline__ void atomAddF(float* p, float v) {
  unsafeAtomicAdd(p, v);  // global_atomic_add_f32
}

// ---------------------------------------------------------------------------
// Fill helpers
// ---------------------------------------------------------------------------
__global__ void fill_f32(float* p, float v, int n) {
  int t = blockIdx.x * blockDim.x + threadIdx.x;
  if (t < n) p[t] = v;
}
__global__ void fill_u32(unsigned* p, unsigned v, int n) {
  int t = blockIdx.x * blockDim.x + threadIdx.x;
  if (t < n) p[t] = v;
}

// ---------------------------------------------------------------------------
// Transpose 8 dense 128x128 weight mats (qW[0..3], aW[0..3]) into [col][k].
// ---------------------------------------------------------------------------
__global__ void transpose_w(const float* __restrict__ qW, const float* __restrict__ aW,
                            float* __restrict__ tQ, float* __restrict__ tA) {
  const int m = blockIdx.x;  // 0..7
  const float* in = (m < 4) ? qW + (size_t)m * HID * HID
                            : aW + (size_t)(m - 4) * HID * HID;
  float* out = (m < 4) ? tQ + (size_t)m * HID * HID
                       : tA + (size_t)(m - 4) * HID * HID;
  for (int idx = threadIdx.x; idx < HID * HID; idx += blockDim.x) {
    int r = idx >> 7, c = idx & 127;
    out[(size_t)c * HID + r] = in[idx];
  }
}

// ---------------------------------------------------------------------------
// Fold a_rel/m_rel into kW/vW, output TRANSPOSED:
//   Wt'[h*32+e, i] = sum_d W[i, h*32+d] * R[h,d,e]
// combo c = ((l*2+t)*2 + m), m=0 -> (kW,a_rel), m=1 -> (vW,m_rel)
// ---------------------------------------------------------------------------
__global__ void fuse_weights(const float* __restrict__ kW, const float* __restrict__ kb,
                             const float* __restrict__ vW, const float* __restrict__ vb,
                             const float* __restrict__ a_rel, const float* __restrict__ m_rel,
                             float* __restrict__ fWt, float* __restrict__ fB) {
  int c = blockIdx.x;      // 0..7
  int lt = c >> 1;         // l*2+t
  int m = c & 1;
  const float* W = (m ? vW : kW) + (size_t)lt * HID * HID;
  const float* B = (m ? vb : kb) + (size_t)lt * HID;
  const float* R = (m ? m_rel : a_rel) + (size_t)lt * 4 * 32 * 32;
  float* oW = fWt + (size_t)c * HID * HID;
  float* oB = fB + (size_t)c * HID;
  for (int idx = threadIdx.x; idx < 129 * 128; idx += blockDim.x) {
    int row = idx >> 7;    // input row i (128 == bias row)
    int col = idx & 127;   // output col h*32+e
    int h = col >> 5, e = col & 31;
    float s = 0.f;
#pragma unroll
    for (int d = 0; d < 32; ++d) {
      float w = (row < 128) ? W[row * HID + h * 32 + d] : B[h * 32 + d];
      s += w * R[(h * 32 + d) * 32 + e];
    }
    if (row < 128) oW[(size_t)col * HID + row] = s;  // transposed store
    else oB[col] = s;
  }
}

// ---------------------------------------------------------------------------
// WMMA GEMM: Y[N,128] = X[N,128] @ W[128,128] + bias, W given transposed
// (Wt[col*128 + k]).  mode 1: Y = relu(g*(X@W+b) + (1-g)*Xold), g=sigmoid(*skipp).
// Block = 256 thr = 8 waves; 32 rows/block; wave w -> col tile w, 2 row tiles.
// A layout (f32 16x4):  lanes 0-15: M=lane, K={k,k+1}; lanes 16-31: K={k+2,k+3}
// B layout (f32 4x16):  lanes 0-15: N=lane, K={k,k+1}; lanes 16-31: K={k+2,k+3}
// C layout: vgpr r -> row r (lanes 0-15) / r+8 (lanes 16-31), col = lane&15.
// ---------------------------------------------------------------------------
__global__ __launch_bounds__(256) void gemm128_wmma(
    const float* __restrict__ X, const float* __restrict__ Wt,
    const float* __restrict__ bias, float* __restrict__ Y,
    int mode, const float* __restrict__ skipp, const float* __restrict__ Xold) {
  __shared__ float Xs[RPB * 132];  // stride 132: bank-conflict-free A loads
  const int rowBase = blockIdx.x * RPB;
  const float* Xrow = X + (size_t)rowBase * HID;
  for (int f = threadIdx.x; f < RPB * HID; f += 256)
    Xs[(f >> 7) * 132 + (f & 127)] = Xrow[f];
  __syncthreads();

  const int lane = threadIdx.x & 31;
  const int colTile = threadIdx.x >> 5;
  const int col = colTile * 16 + (lane & 15);
  const int half = lane >> 4;     // 0: K pair {k,k+1}, 1: K pair {k+2,k+3}
  const int arow = lane & 15;
  const float* wcol = Wt + (size_t)col * HID;

  v8f acc0 = {}, acc1 = {};
#pragma unroll
  for (int k = 0; k < HID; k += 4) {
    const int kk = k + half * 2;
    const v2f b = *(const v2f*)(wcol + kk);   // one global_load_b64
    v2f a0, a1;
    a0.x = Xs[arow * 132 + kk];
    a0.y = Xs[arow * 132 + kk + 1];
    a1.x = Xs[(16 + arow) * 132 + kk];
    a1.y = Xs[(16 + arow) * 132 + kk + 1];
    acc0 = __builtin_amdgcn_wmma_f32_16x16x4_f32(false, a0, false, b, (short)0,
                                                 acc0, false, false);
    acc1 = __builtin_amdgcn_wmma_f32_16x16x4_f32(false, a1, false, b, (short)0,
                                                 acc1, false, false);
  }

  const float bcol = bias[col];
  float g = 0.f;
  if (mode) g = 1.f / (1.f + expf(-skipp[0]));
#pragma unroll
  for (int r = 0; r < 8; ++r) {
#pragma unroll
    for (int tile = 0; tile < 2; ++tile) {
      const int row = rowBase + tile * 16 + r + half * 8;
      const size_t o = (size_t)row * HID + col;
      float y = (tile ? acc1[r] : acc0[r]) + bcol;
      if (mode) {
        y = g * y + (1.f - g) * Xold[o];
        y = fmaxf(y, 0.f);
      }
      Y[o] = y;
    }
  }
}

// ---------------------------------------------------------------------------
// Edge pass 1: attention logits + segment max.  One edge per wave32.
// Lane group of 8 covers one head (8 lanes * float4 = 32 dims).
// ---------------------------------------------------------------------------
__global__ __launch_bounds__(256) void attn_logits(
    const float* __restrict__ q, const float* __restrict__ kr,
    const int* __restrict__ src, const int* __restrict__ dst,
    const float* __restrict__ prel, int E,
    float* __restrict__ alpha, unsigned* __restrict__ amax) {
  const int e = (blockIdx.x * blockDim.x + threadIdx.x) >> 5;
  if (e >= E) return;
  const int lane = threadIdx.x & 31;
  const int s = src[e], d = dst[e];
  const float4 qv = *(const float4*)(q + (size_t)d * HID + lane * 4);
  const float4 kv = *(const float4*)(kr + (size_t)s * HID + lane * 4);
  float p = qv.x * kv.x + qv.y * kv.y + qv.z * kv.z + qv.w * kv.w;
  p += __shfl_xor(p, 1, 32);
  p += __shfl_xor(p, 2, 32);
  p += __shfl_xor(p, 4, 32);
  const int head = lane >> 3;
  if ((lane & 7) == 0) {
    const float aL = p * prel[head] * 0.17677669529663687f;  // 1/sqrt(32)
    alpha[(size_t)e * 4 + head] = aL;
    atomicMax(&amax[(size_t)d * 4 + head], encf(aL));
  }
}

// ---------------------------------------------------------------------------
// Edge pass 2: a = exp(alpha - amax[dst]); denom[dst] += a. One (e,h)/thread.
// ---------------------------------------------------------------------------
__global__ void attn_exp(float* __restrict__ alpha, const int* __restrict__ dst,
                         const unsigned* __restrict__ amax,
                         float* __restrict__ denom, int E) {
  const int t = blockIdx.x * blockDim.x + threadIdx.x;
  if (t >= E * 4) return;
  const int e = t >> 2, h = t & 3;
  const int d = dst[e];
  const float a = expf(alpha[t] - decf(amax[d * 4 + h]));
  alpha[t] = a;
  atomAddF(&denom[d * 4 + h], a);
}

// ---------------------------------------------------------------------------
// Edge pass 3: out[dst] += vr[src] * a/(denom+eps).  One edge per wave32.
// ---------------------------------------------------------------------------
__global__ __launch_bounds__(256) void attn_aggr(
    const float* __restrict__ vr, const float* __restrict__ alpha,
    const float* __restrict__ denom, const int* __restrict__ src,
    const int* __restrict__ dst, int E, float* __restrict__ out) {
  const int e = (blockIdx.x * blockDim.x + threadIdx.x) >> 5;
  if (e >= E) return;
  const int lane = threadIdx.x & 31;
  const int s = src[e], d = dst[e];
  const int head = lane >> 3;
  const float a = alpha[(size_t)e * 4 + head];
  const float w = a / (denom[(size_t)d * 4 + head] + 1e-16f);
  const float4 vv = *(const float4*)(vr + (size_t)s * HID + lane * 4);
  float* o = out + (size_t)d * HID + lane * 4;
  atomAddF(o + 0, vv.x * w);
  atomAddF(o + 1, vv.y * w);
  atomAddF(o + 2, vv.z * w);
  atomAddF(o + 3, vv.w * w);
}

// exact gelu, in place
__global__ void gelu_inplace(float* __restrict__ x, int n) {
  const int t = blockIdx.x * blockDim.x + threadIdx.x;
  if (t < n) {
    const float v = x[t];
    x[t] = 0.5f * v * (1.f + erff(v * 0.70710678118654752f));
  }
}

// column mean of xg: one block per column
__global__ __launch_bounds__(256) void col_mean(const float* __restrict__ xg,
                                                int N, float* __restrict__ gmean) {
  __shared__ float red[256];
  const int col = blockIdx.x, t = threadIdx.x;
  float s = 0.f;
  for (int r = t; r < N; r += 256) s += xg[(size_t)r * HID + col];
  red[t] = s;
  __syncthreads();
  for (int sft = 128; sft > 0; sft >>= 1) {
    if (t < sft) red[t] += red[t + sft];
    __syncthreads();
  }
  if (t == 0) gmean[col] = red[0] / (float)N;
}

// predictor MLP: one block (128 thr) per batch element
__global__ __launch_bounds__(128) void predictor(
    const float* __restrict__ xd, const float* __restrict__ gmean,
    const int* __restrict__ drug_idx, const float* __restrict__ pW1,
    const float* __restrict__ pb1, const float* __restrict__ pW2,
    const float* __restrict__ pb2, float* __restrict__ out) {
  __shared__ float comb[256];
  __shared__ float red[128];
  const int b = blockIdx.x, t = threadIdx.x;
  const int r = drug_idx[b];
  comb[t] = xd[(size_t)r * HID + t];
  comb[128 + t] = gmean[t];
  __syncthreads();
  float h = pb1[t];
#pragma unroll 8
  for (int i = 0; i < 256; ++i) h += comb[i] * pW1[i * HID + t];
  h = fmaxf(h, 0.f);
  red[t] = h * pW2[t];
  __syncthreads();
  for (int sft = 64; sft > 0; sft >>= 1) {
    if (t < sft) red[t] += red[t + sft];
    __syncthreads();
  }
  if (t == 0) out[b] = red[0] + pb2[0];
}

// ---------------------------------------------------------------------------
extern "C" void kernel_launch(void* const* d_in, const int* in_sizes, int n_in,
                              void* d_out, int out_size, void* d_ws, size_t ws_size,
                              hipStream_t stream) {
  const float* x_drug = (const float*)d_in[0];
  const float* x_gene = (const float*)d_in[1];
  const float* kW = (const float*)d_in[2];
  const float* kb = (const float*)d_in[3];
  const float* qW = (const float*)d_in[4];
  const float* qb = (const float*)d_in[5];
  const float* vW = (const float*)d_in[6];
  const float* vb = (const float*)d_in[7];
  const float* aW = (const float*)d_in[8];
  const float* ab = (const float*)d_in[9];
  const float* skip = (const float*)d_in[10];
  const float* a_rel = (const float*)d_in[11];
  const float* m_rel = (const float*)d_in[12];
  const float* p_rel = (const float*)d_in[13];
  const float* pW1 = (const float*)d_in[14];
  const float* pb1 = (const float*)d_in[15];
  const float* pW2 = (const float*)d_in[16];
  const float* pb2 = (const float*)d_in[17];
  const int* dg_src = (const int*)d_in[18];
  const int* dg_dst = (const int*)d_in[19];
  const int* gd_src = (const int*)d_in[20];
  const int* gd_dst = (const int*)d_in[21];
  const int* drug_idx = (const int*)d_in[22];
  const int E = in_sizes[18];
  const int BATCH = in_sizes[22];

  // ---- workspace layout (floats) ----
  float* ws = (float*)d_ws;
  size_t off = 0;
  auto alloc = [&](size_t n) { float* p = ws + off; off += n; return p; };
  float* xd   = alloc((size_t)ND * HID);
  float* xg   = alloc((size_t)NG * HID);
  float* qd   = alloc((size_t)ND * HID);
  float* krD  = alloc((size_t)ND * HID);
  float* vrD  = alloc((size_t)ND * HID);
  float* qg   = alloc((size_t)NG * HID);
  float* krG  = alloc((size_t)NG * HID);
  float* vrG  = alloc((size_t)NG * HID);
  float* aggD = alloc((size_t)ND * HID);
  float* aggG = alloc((size_t)NG * HID);
  float* alpha0 = alloc((size_t)E * 4);
  float* alpha1 = alloc((size_t)E * 4);
  unsigned* amax = (unsigned*)alloc((size_t)NG * 4);
  float* denom = alloc((size_t)NG * 4);
  float* fWt = alloc(8 * (size_t)HID * HID);   // fused k/v weights, transposed
  float* fB  = alloc(8 * (size_t)HID);
  float* tQ  = alloc(4 * (size_t)HID * HID);   // qW transposed
  float* tA  = alloc(4 * (size_t)HID * HID);   // aW transposed
  float* gmean = alloc(HID);

  hipMemcpyAsync(xd, x_drug, (size_t)ND * HID * sizeof(float),
                 hipMemcpyDeviceToDevice, stream);
  hipMemcpyAsync(xg, x_gene, (size_t)NG * HID * sizeof(float),
                 hipMemcpyDeviceToDevice, stream);

  fuse_weights<<<8, 256, 0, stream>>>(kW, kb, vW, vb, a_rel, m_rel, fWt, fB);
  transpose_w<<<8, 256, 0, stream>>>(qW, aW, tQ, tA);

  const int edgeWaveBlocks = (E + 7) / 8;      // 8 edges per 256-thr block
  const int expBlocks = (E * 4 + 255) / 256;

  for (int l = 0; l < LAYERS; ++l) {
    const int ltD = l * 2 + 0, ltG = l * 2 + 1;
    // ---- projections (WMMA GEMMs); kr/vr use relation-fused weights ----
    gemm128_wmma<<<ND / RPB, 256, 0, stream>>>(xd, tQ + (size_t)ltD * HID * HID,
        qb + (size_t)ltD * HID, qd, 0, nullptr, nullptr);
    gemm128_wmma<<<ND / RPB, 256, 0, stream>>>(xd, fWt + (size_t)(ltD * 2 + 0) * HID * HID,
        fB + (size_t)(ltD * 2 + 0) * HID, krD, 0, nullptr, nullptr);
    gemm128_wmma<<<ND / RPB, 256, 0, stream>>>(xd, fWt + (size_t)(ltD * 2 + 1) * HID * HID,
        fB + (size_t)(ltD * 2 + 1) * HID, vrD, 0, nullptr, nullptr);
    gemm128_wmma<<<NG / RPB, 256, 0, stream>>>(xg, tQ + (size_t)ltG * HID * HID,
        qb + (size_t)ltG * HID, qg, 0, nullptr, nullptr);
    gemm128_wmma<<<NG / RPB, 256, 0, stream>>>(xg, fWt + (size_t)(ltG * 2 + 0) * HID * HID,
        fB + (size_t)(ltG * 2 + 0) * HID, krG, 0, nullptr, nullptr);
    gemm128_wmma<<<NG / RPB, 256, 0, stream>>>(xg, fWt + (size_t)(ltG * 2 + 1) * HID * HID,
        fB + (size_t)(ltG * 2 + 1) * HID, vrG, 0, nullptr, nullptr);

    // ---- edge type 0: drug -> gene (dst = genes) ----
    fill_u32<<<(NG * 4 + 255) / 256, 256, 0, stream>>>(amax, ENC_NEGINF, NG * 4);
    fill_f32<<<(NG * 4 + 255) / 256, 256, 0, stream>>>(denom, 0.f, NG * 4);
    fill_f32<<<(NG * HID + 255) / 256, 256, 0, stream>>>(aggG, 0.f, NG * HID);
    attn_logits<<<edgeWaveBlocks, 256, 0, stream>>>(qg, krD, dg_src, dg_dst,
        p_rel + (size_t)ltD * 4, E, alpha0, amax);
    attn_exp<<<expBlocks, 256, 0, stream>>>(alpha0, dg_dst, amax, denom, E);
    attn_aggr<<<edgeWaveBlocks, 256, 0, stream>>>(vrD, alpha0, denom, dg_src,
        dg_dst, E, aggG);

    // ---- edge type 1: gene -> drug (dst = drugs) ----
    fill_u32<<<(ND * 4 + 255) / 256, 256, 0, stream>>>(amax, ENC_NEGINF, ND * 4);
    fill_f32<<<(ND * 4 + 255) / 256, 256, 0, stream>>>(denom, 0.f, ND * 4);
    fill_f32<<<(ND * HID + 255) / 256, 256, 0, stream>>>(aggD, 0.f, ND * HID);
    attn_logits<<<edgeWaveBlocks, 256, 0, stream>>>(qd, krG, gd_src, gd_dst,
        p_rel + (size_t)ltG * 4, E, alpha1, amax);
    attn_exp<<<expBlocks, 256, 0, stream>>>(alpha1, gd_dst, amax, denom, E);
    attn_aggr<<<edgeWaveBlocks, 256, 0, stream>>>(vrG, alpha1, denom, gd_src,
        gd_dst, E, aggD);

    // ---- finish: gelu -> aW GEMM with fused sigmoid-gated skip + relu ----
    gelu_inplace<<<(ND * HID + 255) / 256, 256, 0, stream>>>(aggD, ND * HID);
    gelu_inplace<<<(NG * HID + 255) / 256, 256, 0, stream>>>(aggG, NG * HID);
    gemm128_wmma<<<ND / RPB, 256, 0, stream>>>(aggD, tA + (size_t)ltD * HID * HID,
        ab + (size_t)ltD * HID, xd, 1, skip + ltD, xd);
    gemm128_wmma<<<NG / RPB, 256, 0, stream>>>(aggG, tA + (size_t)ltG * HID * HID,
        ab + (size_t)ltG * HID, xg, 1, skip + ltG, xg);
  }

  col_mean<<<HID, 256, 0, stream>>>(xg, NG, gmean);
  predictor<<<BATCH, 128, 0, stream>>>(xd, gmean, drug_idx, pW1, pb1, pW2, pb2,
                                       (float*)d_out);
}